// RNNTLoss_62251255988521
// MI455X (gfx1250) — compile-verified
//
#include <hip/hip_runtime.h>
#include <stdint.h>

// Problem constants (match reference: B=8, T=256, U=65, V1=1024)
#define BB 8
#define TT 256
#define UU 65
#define VV 1024
#define BIG_NEG (-1.0e30f)

// s_blank: TT*UU floats at LDS offset 0, s_emit: TT*(UU-1) floats after it
#define N_BLANK (TT * UU)        // 16640
#define N_EMIT  (TT * (UU - 1))  // 16384
#define SMEM_BYTES ((N_BLANK + N_EMIT) * sizeof(float))  // 132096 B < 320KB/WGP

typedef unsigned int v4u __attribute__((ext_vector_type(4)));
typedef int          v4i __attribute__((ext_vector_type(4)));
typedef int          v8i __attribute__((ext_vector_type(8)));

__device__ __forceinline__ float logaddexpf_(float x, float y) {
    float m = fmaxf(x, y);
    float d = fabsf(x - y);
    return m + log1pf(expf(-d));
}

// CDNA5 async gather: 4B global -> LDS, no VGPR writeback, tracked by ASYNCcnt.
__device__ __forceinline__ void async_gather_f32(unsigned lds_byte_addr, const float* g) {
    asm volatile("global_load_async_to_lds_b32 %0, %1, off"
                 :: "v"(lds_byte_addr), "v"(g)
                 : "memory");
}

__device__ __forceinline__ unsigned lds_addr_of(const void* p) {
    // Flat LDS pointers carry the LDS byte address in the low 32 bits.
    return (unsigned)(uintptr_t)p;
}

extern "C" __global__ void __launch_bounds__(256)
rnnt_forward_kernel(const float* __restrict__ acts,
                    const long long* __restrict__ labels,
                    const long long* __restrict__ act_lens,
                    const long long* __restrict__ label_lens,
                    float* __restrict__ ws) {
    extern __shared__ float smem[];
    float* s_blank = smem;            // [TT][UU]   at LDS byte offset 0
    float* s_emit  = smem + N_BLANK;  // [TT][UU-1]

    const int b   = blockIdx.x;
    const int tid = threadIdx.x;
    const float* actb = acts + (size_t)b * TT * UU * VV;

    // ------------- Phase 1a: blank plane via Tensor Data Mover (one DMA) -------------
    // blank[t][u] = acts[b,t,u,0]: a 16640-row x 1-elem tensor with row stride 1024.
#if __has_builtin(__builtin_amdgcn_tensor_load_to_lds)
    if (tid < 32) {  // wave 0 issues the single TDM descriptor
        unsigned long long ga = (unsigned long long)(uintptr_t)actb;
        v4u g0;
        g0.x = 1u;                                   // count=1 (valid), no gather mode
        g0.y = 0u;                                   // lds_addr = 0 (s_blank base)
        g0.z = (unsigned)ga;                         // global_addr[31:0]
        g0.w = (unsigned)((ga >> 32) & 0x01FFFFFFu)  // global_addr[56:32]
             | (2u << 30);                           // type = 2 ("image")
        v8i g1;
        g1[0] = 0x20000;                             // data_size=2 -> 4 bytes; mask=0
        g1[1] = (int)(1u << 16);                     // tensor_dim0 = 1 (lo16)
        g1[2] = (int)((unsigned)N_BLANK << 16);      // tensor_dim1 lo16; dim0 hi16=0
        g1[3] = (int)((N_BLANK >> 16) | (1u << 16)); // tensor_dim1 hi16 | tile_dim0=1
        g1[4] = N_BLANK;                             // tile_dim1 = 16640; tile_dim2 = 0
        g1[5] = VV;                                  // tensor_dim0_stride = 1024 elems
        g1[6] = 0;                                   // stride hi16 | dim1_stride lo16
        g1[7] = 0;
        v4i z4 = {0, 0, 0, 0};
#if __clang_major__ >= 23
        v8i z8 = {0, 0, 0, 0, 0, 0, 0, 0};
        __builtin_amdgcn_tensor_load_to_lds(g0, g1, z4, z4, z8, 0);
#else
        __builtin_amdgcn_tensor_load_to_lds(g0, g1, z4, z4, 0);
#endif
    }
#else
    for (int i = tid; i < N_BLANK; i += 256) {
        const float* g = actb + (size_t)i * VV;
        async_gather_f32(lds_addr_of(&s_blank[i]), g);
    }
#endif

    // ------------- Phase 1b: emit gather via per-lane async loads -------------
    // emit[t][u] = acts[b,t,u,lab[u]]; u = j & 63 is fixed per thread (256 % 64 == 0).
    {
        const int u   = tid & 63;
        const int lab = (int)labels[b * (UU - 1) + u];
        for (int j = tid; j < N_EMIT; j += 256) {
            int t = j >> 6;
            const float* g = actb + ((size_t)(t * UU + u) * VV + lab);
            async_gather_f32(lds_addr_of(&s_emit[j]), g);
        }
    }

    asm volatile("s_wait_asynccnt 0" ::: "memory");
#if __has_builtin(__builtin_amdgcn_tensor_load_to_lds)
    if (tid < 32) {
#if __has_builtin(__builtin_amdgcn_s_wait_tensorcnt)
        __builtin_amdgcn_s_wait_tensorcnt(0);
#else
        asm volatile("s_wait_tensorcnt 0x0" ::: "memory");
#endif
    }
#endif
    __syncthreads();

    // ---------------- Phase 2: single-wave anti-diagonal DP ----------------
    if (tid >= 32) return;
    const int lane = tid;

    const int flen = (int)act_lens[b];    // = TT in harness
    const int glen = (int)label_lens[b];  // = UU-1 in harness

    // Lane l owns u0 = l, u1 = l + 32; u2 = 64 computed redundantly by all lanes.
    const int u0 = lane;
    const int u1 = lane + 32;   // 32..63
    const int u2 = UU - 1;      // 64

    float a0 = (lane == 0) ? 0.0f : BIG_NEG;  // alpha[0][0] = 0
    float a1 = BIG_NEG;
    float a2 = BIG_NEG;

    float sv0 = BIG_NEG, sv1 = BIG_NEG, sv2 = BIG_NEG;  // saved alpha[flen-1][glen]
    if (flen == 1 && glen == 0 && lane == 0) sv0 = 0.0f;

    for (int d = 1; d <= TT + UU - 2; ++d) {  // diagonals 1..319
        // Read neighbors from the PREVIOUS diagonal (pre-update values).
        float l0 = __shfl_up(a0, 1, 32);          // A_prev[u0-1]
        float l1 = __shfl_up(a1, 1, 32);          // A_prev[u1-1] (lanes >= 1)
        float a0_31 = __shfl(a0, 31, 32);         // A_prev[31]
        float a1_31 = __shfl(a1, 31, 32);         // A_prev[63]
        if (lane == 0) l1 = a0_31;                // u=32 neighbor is u=31
        float l2 = a1_31;                         // u=64 neighbor is u=63

        // slot 0: u = u0, t = d - u0
        float n0 = a0;
        {
            int t = d - u0;
            if (t >= 0 && t < TT) {
                float x = (t >= 1) ? a0 + s_blank[(t - 1) * UU + u0] : BIG_NEG;
                float y = (u0 >= 1) ? l0 + s_emit[t * (UU - 1) + (u0 - 1)] : BIG_NEG;
                n0 = logaddexpf_(x, y);
                if (t == flen - 1 && u0 == glen) sv0 = n0;
            }
        }
        // slot 1: u = u1 (>= 32, so emit term always present)
        float n1 = a1;
        {
            int t = d - u1;
            if (t >= 0 && t < TT) {
                float x = (t >= 1) ? a1 + s_blank[(t - 1) * UU + u1] : BIG_NEG;
                float y = l1 + s_emit[t * (UU - 1) + (u1 - 1)];
                n1 = logaddexpf_(x, y);
                if (t == flen - 1 && u1 == glen) sv1 = n1;
            }
        }
        // slot 2: u = 64 (identical on every lane; inputs are wave-uniform)
        float n2 = a2;
        {
            int t = d - u2;
            if (t >= 0 && t < TT) {
                float x = (t >= 1) ? a2 + s_blank[(t - 1) * UU + u2] : BIG_NEG;
                float y = l2 + s_emit[t * (UU - 1) + (u2 - 1)];
                n2 = logaddexpf_(x, y);
                if (t == flen - 1 && u2 == glen) sv2 = n2;
            }
        }
        a0 = n0; a1 = n1; a2 = n2;
    }

    // Extract alpha[flen-1][glen] from the owning slot/lane.
    float alpha_end;
    if (glen < 32) {
        alpha_end = __shfl(sv0, glen, 32);
    } else if (glen < 64) {
        alpha_end = __shfl(sv1, glen - 32, 32);
    } else {
        alpha_end = sv2;  // wave-uniform
    }

    if (lane == 0) {
        float ll = alpha_end + s_blank[(flen - 1) * UU + glen];
        ws[b] = -ll;
    }
}

extern "C" __global__ void rnnt_reduce_kernel(const float* __restrict__ ws,
                                              float* __restrict__ out) {
    if (threadIdx.x == 0 && blockIdx.x == 0) {
        float s = 0.0f;
        for (int i = 0; i < BB; ++i) s += ws[i];  // fixed order: deterministic
        out[0] = s;
    }
}

extern "C" void kernel_launch(void* const* d_in, const int* in_sizes, int n_in,
                              void* d_out, int out_size, void* d_ws, size_t ws_size,
                              hipStream_t stream) {
    const float*     acts       = (const float*)d_in[0];
    const long long* labels     = (const long long*)d_in[1];
    const long long* act_lens   = (const long long*)d_in[2];
    const long long* label_lens = (const long long*)d_in[3];
    float* ws  = (float*)d_ws;
    float* out = (float*)d_out;

    rnnt_forward_kernel<<<dim3(BB), dim3(256), SMEM_BYTES, stream>>>(
        acts, labels, act_lens, label_lens, ws);
    rnnt_reduce_kernel<<<dim3(1), dim3(32), 0, stream>>>(ws, out);
}